// GATv2_33784212750631
// MI455X (gfx1250) — compile-verified
//
#include <hip/hip_runtime.h>
#include <hip/hip_bf16.h>

typedef __attribute__((ext_vector_type(2))) float v2f;
typedef __attribute__((ext_vector_type(8))) float v8f;

#define CH 128

// ---------------------------------------------------------------------------
// Kernel A: h = node_feat[N,118] @ W_node[118,128] + b_node   (f32 WMMA)
// One wave per 16x16 output tile; 8 waves/block cover all 8 column tiles.
// WMMA f32 16x16x4 fragment layouts (ISA 7.12.2):
//   A 16x4 : lane<16 -> M=lane,    K = {0,1} in v[0],v[1]; lane>=16 -> K={2,3}
//   B 4x16 : lane%16 -> N,         lane/16 selects K-half, v[i] -> K within half
//   C 16x16: v[i] -> M = (lane/16)*8 + i, N = lane%16
// ---------------------------------------------------------------------------
__global__ void node_linear_wmma(const float* __restrict__ A,   // [N,118]
                                 const float* __restrict__ W,   // [118,128]
                                 const float* __restrict__ bias,// [128]
                                 float* __restrict__ H,         // [N,128]
                                 int N) {
  const int K = 118;
  const int m0    = blockIdx.x * 16;
  const int wave  = threadIdx.x >> 5;       // 0..7 -> column tile
  const int n0    = wave * 16;
  const int lane  = threadIdx.x & 31;
  const int row   = lane & 15;              // M for A-frag, N for B/C-frag
  const int khalf = lane >> 4;

  int r = m0 + row; if (r >= N) r = N - 1;  // clamp (N%16==0 anyway)

  v8f acc = {};
  for (int k0 = 0; k0 < K; k0 += 4) {
    const int ka = k0 + khalf * 2;
    v2f a, b;
    a.x = (ka     < K) ? A[(size_t)r * K + ka]       : 0.0f;
    a.y = (ka + 1 < K) ? A[(size_t)r * K + ka + 1]   : 0.0f;
    b.x = (ka     < K) ? W[(size_t)ka * CH + n0 + row]       : 0.0f;
    b.y = (ka + 1 < K) ? W[(size_t)(ka + 1) * CH + n0 + row] : 0.0f;
    acc = __builtin_amdgcn_wmma_f32_16x16x4_f32(false, a, false, b,
                                                (short)0, acc, false, false);
  }

  const float bn = bias[n0 + row];
#pragma unroll
  for (int v = 0; v < 8; ++v) {
    const int m = m0 + khalf * 8 + v;
    if (m < N) H[(size_t)m * CH + n0 + row] = acc[v] + bn;
  }
}

// ---------------------------------------------------------------------------
// Kernel B: a_src = h @ W_a1[0:128,:], a_dst = h @ W_a1[128:256,:]  (f32 WMMA)
// 16 waves/block: waves 0..7 -> a_src column tiles, 8..15 -> a_dst.
// ---------------------------------------------------------------------------
__global__ void attn_proj_wmma(const float* __restrict__ H,    // [N,128]
                               const float* __restrict__ Wa1,  // [256,128]
                               float* __restrict__ Asrc,       // [N,128]
                               float* __restrict__ Adst,       // [N,128]
                               int N) {
  const int K = 128;
  const int m0    = blockIdx.x * 16;
  const int j     = threadIdx.x >> 5;       // 0..15
  const int lane  = threadIdx.x & 31;
  const int row   = lane & 15;
  const int khalf = lane >> 4;

  const bool dstHalf = (j >= 8);
  const float* Wb = Wa1 + (dstHalf ? (size_t)128 * CH : 0);
  const int n0 = (dstHalf ? j - 8 : j) * 16;

  int r = m0 + row; if (r >= N) r = N - 1;

  v8f acc = {};
  for (int k0 = 0; k0 < K; k0 += 4) {
    const int ka = k0 + khalf * 2;
    v2f a, b;
    a.x = H[(size_t)r * K + ka];
    a.y = H[(size_t)r * K + ka + 1];
    b.x = Wb[(size_t)ka * CH + n0 + row];
    b.y = Wb[(size_t)(ka + 1) * CH + n0 + row];
    acc = __builtin_amdgcn_wmma_f32_16x16x4_f32(false, a, false, b,
                                                (short)0, acc, false, false);
  }

  float* Out = dstHalf ? Adst : Asrc;
#pragma unroll
  for (int v = 0; v < 8; ++v) {
    const int m = m0 + khalf * 8 + v;
    if (m < N) Out[(size_t)m * CH + n0 + row] = acc[v];
  }
}

// ---------------------------------------------------------------------------
// Kernel C: per edge  e = dot(leaky_relu(a_src[s] + a_dst[d] + b_a1), W_a2) + b_a2
// One wave (32 lanes) per edge; each lane owns 4 channels (float4 loads ->
// 512B contiguous burst per table per wave). wave32 shfl_xor reduction.
// ---------------------------------------------------------------------------
__global__ void edge_attn(const float* __restrict__ Asrc,  // [N,128]
                          const float* __restrict__ Adst,  // [N,128]
                          const int*   __restrict__ src,   // [E]
                          const int*   __restrict__ dst,   // [E]
                          const float* __restrict__ ba1,   // [128]
                          const float* __restrict__ Wa2,   // [128]
                          const float* __restrict__ ba2,   // [1]
                          float* __restrict__ out,         // [E]
                          int E) {
  const int gid  = blockIdx.x * blockDim.x + threadIdx.x;
  const int e    = gid >> 5;
  const int lane = gid & 31;
  if (e >= E) return;

  const int s = src[e];
  const int d = dst[e];

  const float4 xs = ((const float4*)(Asrc + (size_t)s * CH))[lane];
  const float4 xd = ((const float4*)(Adst + (size_t)d * CH))[lane];
  const float4 bb = ((const float4*)ba1)[lane];
  const float4 ww = ((const float4*)Wa2)[lane];

  float v0 = xs.x + xd.x + bb.x;
  float v1 = xs.y + xd.y + bb.y;
  float v2 = xs.z + xd.z + bb.z;
  float v3 = xs.w + xd.w + bb.w;
  v0 = v0 > 0.0f ? v0 : 0.01f * v0;
  v1 = v1 > 0.0f ? v1 : 0.01f * v1;
  v2 = v2 > 0.0f ? v2 : 0.01f * v2;
  v3 = v3 > 0.0f ? v3 : 0.01f * v3;

  float acc = v0 * ww.x + v1 * ww.y + v2 * ww.z + v3 * ww.w;

#pragma unroll
  for (int off = 16; off > 0; off >>= 1)
    acc += __shfl_xor(acc, off, 32);

  if (lane == 0) out[e] = acc + ba2[0];
}

// ---------------------------------------------------------------------------
// Host-side launcher.
// Inputs (setup_inputs order):
//  0 node_feat [N,118] f32   4 W_node [118,128]  8 W_a1 [256,128]  12 layer_num
//  1 edge_feat [E,22]  f32   5 b_node [128]      9 b_a1 [128]
//  2 src [E] i32             6 W_edge [22,128]  10 W_a2 [128,1]
//  3 dst [E] i32             7 b_edge [128]     11 b_a2 [1]
// Note: edge_feat/W_edge/b_edge are dead code in the reference (eh unused),
// and the layer loop is idempotent (h never changes) -> single pass.
// ---------------------------------------------------------------------------
extern "C" void kernel_launch(void* const* d_in, const int* in_sizes, int n_in,
                              void* d_out, int out_size, void* d_ws, size_t ws_size,
                              hipStream_t stream) {
  const float* node_feat = (const float*)d_in[0];
  const int*   src       = (const int*)d_in[2];
  const int*   dst       = (const int*)d_in[3];
  const float* W_node    = (const float*)d_in[4];
  const float* b_node    = (const float*)d_in[5];
  const float* W_a1      = (const float*)d_in[8];
  const float* b_a1      = (const float*)d_in[9];
  const float* W_a2      = (const float*)d_in[10];
  const float* b_a2      = (const float*)d_in[11];

  const int N = in_sizes[0] / 118;   // 10000
  const int E = in_sizes[2];         // 640000

  float* h     = (float*)d_ws;                     // [N,128]  5.12 MB
  float* a_src = h + (size_t)N * CH;               // [N,128]  5.12 MB
  float* a_dst = a_src + (size_t)N * CH;           // [N,128]  5.12 MB

  const int mtiles = (N + 15) / 16;                // 625

  // A: node linear (WMMA f32). 8 waves/block = 8 column tiles of CH=128.
  node_linear_wmma<<<dim3(mtiles), dim3(256), 0, stream>>>(
      node_feat, W_node, b_node, h, N);

  // B: per-node src/dst attention projections (WMMA f32). 16 waves/block.
  attn_proj_wmma<<<dim3(mtiles), dim3(512), 0, stream>>>(
      h, W_a1, a_src, a_dst, N);

  // C: edge pass, wave-per-edge. 8 edges per 256-thread block.
  const int blocks = (E * 32 + 255) / 256;
  edge_attn<<<dim3(blocks), dim3(256), 0, stream>>>(
      a_src, a_dst, src, dst, b_a1, W_a2, b_a2, (float*)d_out, E);
}